// GraphAttentionNetwork_80444737454870
// MI455X (gfx1250) — compile-verified
//
#include <hip/hip_runtime.h>
#include <cmath>

#define NN 100000
#define EE 3200000
#define NEGS 0.2f

typedef __attribute__((ext_vector_type(2))) float v2f;
typedef __attribute__((ext_vector_type(8))) float v8f;

// ---- order-preserving float<->uint mapping for atomic segment-max ----
__device__ __forceinline__ unsigned f2mono(float f) {
    unsigned u = __float_as_uint(f);
    return (u & 0x80000000u) ? ~u : (u | 0x80000000u);
}
__device__ __forceinline__ float mono2f(unsigned u) {
    return (u & 0x80000000u) ? __uint_as_float(u ^ 0x80000000u) : __uint_as_float(~u);
}
#define MONO_NEG_INF 0x007FFFFFu   // f2mono(-inf)

// ---------------------------------------------------------------------
// Tiny weight prep: Wcat1[16x128] = concat_h W1[h] ; wea1[h][16]=We1[h]@a_e1[h];
// wea2[16] = We2 @ a_e2
__global__ void prep_weights(const float* __restrict__ W1, const float* __restrict__ We1,
                             const float* __restrict__ att1, const float* __restrict__ We2,
                             const float* __restrict__ att2,
                             float* __restrict__ Wcat1, float* __restrict__ wea1,
                             float* __restrict__ wea2)
{
    int tid = threadIdx.x;
    for (int i = tid; i < 2048; i += blockDim.x) {
        int k = i >> 7, c = i & 127, h = c >> 5, o = c & 31;
        Wcat1[i] = W1[h * 512 + k * 32 + o];
    }
    for (int i = tid; i < 64; i += blockDim.x) {
        int h = i >> 4, k = i & 15;
        float s = 0.f;
        for (int o = 0; o < 32; ++o) s += We1[h * 512 + k * 32 + o] * att1[h * 96 + 64 + o];
        wea1[i] = s;
    }
    for (int i = tid; i < 16; i += blockDim.x) {
        float s = 0.f;
        for (int o = 0; o < 32; ++o) s += We2[i * 32 + o] * att2[64 + o];
        wea2[i] = s;
    }
}

// ---------------------------------------------------------------------
__global__ void init_buffers(unsigned* __restrict__ emax1, float* __restrict__ denom1,
                             float* __restrict__ acc1, unsigned* __restrict__ emax2,
                             float* __restrict__ denom2, float* __restrict__ acc2)
{
    int i0 = blockIdx.x * blockDim.x + threadIdx.x;
    int stride = gridDim.x * blockDim.x;
    for (int i = i0; i < NN * 128; i += stride) acc1[i] = 0.f;
    for (int i = i0; i < NN * 32; i += stride) acc2[i] = 0.f;
    for (int i = i0; i < NN * 4; i += stride) { emax1[i] = MONO_NEG_INF; denom1[i] = 0.f; }
    for (int i = i0; i < NN; i += stride) { emax2[i] = MONO_NEG_INF; denom2[i] = 0.f; }
}

// ---------------------------------------------------------------------
// FP32 WMMA GEMM: C[MxNcols] = A[MxK] * B[KxNcols] (+bias, nelu x ELU)
// One wave32 per 16x16 output tile, V_WMMA_F32_16X16X4_F32, K stepped by 4.
// A lane layout: lane l holds A[l%16][kk], A[l%16][kk+1], kk = 2*(l/16).
// B lane layout: lane l holds B[kk][l%16], B[kk+1][l%16].
// D lane layout: lane l, vgpr v -> row v + 8*(l/16), col l%16.
__global__ void wmma_gemm(const float* __restrict__ A, int lda,
                          const float* __restrict__ B, int ldb,
                          float* __restrict__ C, int ldc,
                          int M, int Ncols, int K,
                          const float* __restrict__ bias, int nelu)
{
    int lane = threadIdx.x & 31;
    int tile = blockIdx.x * (blockDim.x >> 5) + (threadIdx.x >> 5);
    int tilesN = Ncols >> 4;
    int total = (M >> 4) * tilesN;
    if (tile >= total) return;                 // wave-uniform: EXEC stays all-ones
    int tm = tile / tilesN, tn = tile % tilesN;
    int r = lane & 15;
    int kk = (lane >> 4) << 1;
    const float* Ap = A + (size_t)(tm * 16 + r) * lda + kk;
    const float* Bp = B + (size_t)kk * ldb + (tn * 16 + r);
    v8f acc = {};
    for (int k0 = 0; k0 < K; k0 += 4) {
        v2f a; a.x = Ap[0]; a.y = Ap[1];
        v2f b; b.x = Bp[0]; b.y = Bp[ldb];
        acc = __builtin_amdgcn_wmma_f32_16x16x4_f32(false, a, false, b,
                                                    (short)0, acc, false, false);
        Ap += 4;
        Bp += (size_t)ldb * 4;
    }
    int colg = tn * 16 + r;
    float bv = bias ? bias[colg] : 0.f;
    int rbase = tm * 16 + ((lane >> 4) << 3);
#pragma unroll
    for (int v = 0; v < 8; ++v) {
        float xv = acc[v] + bv;
        for (int t = 0; t < nelu; ++t) xv = (xv > 0.f) ? xv : (expf(xv) - 1.f);
        C[(size_t)(rbase + v) * ldc + colg] = xv;
    }
}

// ---------------------------------------------------------------------
// s_i/s_j per (node, head) for layer 1 (32-dot against att1 slices)
__global__ void scores1_kernel(const float* __restrict__ xw1, const float* __restrict__ att1,
                               float* __restrict__ s_i1, float* __restrict__ s_j1)
{
    int t = blockIdx.x * blockDim.x + threadIdx.x;
    if (t >= NN * 4) return;
    int n = t >> 2, h = t & 3;
    const float* a = att1 + h * 96;
    const float* xp = xw1 + (size_t)n * 128 + h * 32;
    float si = 0.f, sj = 0.f;
#pragma unroll
    for (int o = 0; o < 32; ++o) { float v = xp[o]; si += v * a[o]; sj += v * a[32 + o]; }
    s_i1[t] = si; s_j1[t] = sj;
}

__global__ void scores2_kernel(const float* __restrict__ hw2, const float* __restrict__ att2,
                               float* __restrict__ s_i2, float* __restrict__ s_j2)
{
    int n = blockIdx.x * blockDim.x + threadIdx.x;
    if (n >= NN) return;
    const float* xp = hw2 + (size_t)n * 32;
    float si = 0.f, sj = 0.f;
#pragma unroll
    for (int o = 0; o < 32; ++o) { float v = xp[o]; si += v * att2[o]; sj += v * att2[32 + o]; }
    s_i2[n] = si; s_j2[n] = sj;
}

// ---------------------------------------------------------------------
// Edge pass 1 (layer 1): thread per (edge, head) -> leaky score -> atomic segment-max
__global__ void edge_scores_l1(const int* __restrict__ src, const int* __restrict__ dst,
                               const float* __restrict__ eattr, const float* __restrict__ wea1,
                               const float* __restrict__ s_i1, const float* __restrict__ s_j1,
                               unsigned* __restrict__ emax1)
{
    int t = blockIdx.x * blockDim.x + threadIdx.x;
    if (t >= EE * 4) return;
    int e = t >> 2, h = t & 3;
    int s = src[e], d = dst[e];
    float se = 0.f;
#pragma unroll
    for (int k = 0; k < 16; ++k) se += eattr[(size_t)e * 16 + k] * wea1[h * 16 + k];
    float sc = s_i1[d * 4 + h] + s_j1[s * 4 + h] + se;
    sc = (sc >= 0.f) ? sc : NEGS * sc;
    atomicMax(&emax1[d * 4 + h], f2mono(sc));
}

__global__ void edge_scores_l2(const int* __restrict__ src, const int* __restrict__ dst,
                               const float* __restrict__ eattr, const float* __restrict__ wea2,
                               const float* __restrict__ s_i2, const float* __restrict__ s_j2,
                               unsigned* __restrict__ emax2)
{
    int e = blockIdx.x * blockDim.x + threadIdx.x;
    if (e >= EE) return;
    int s = src[e], d = dst[e];
    float se = 0.f;
#pragma unroll
    for (int k = 0; k < 16; ++k) se += eattr[(size_t)e * 16 + k] * wea2[k];
    float sc = s_i2[d] + s_j2[s] + se;
    sc = (sc >= 0.f) ? sc : NEGS * sc;
    atomicMax(&emax2[d], f2mono(sc));
}

// decode monotonic-uint max back to float; non-finite (no-edge nodes) -> 0
__global__ void decode_emax(unsigned* __restrict__ em, int n)
{
    int i = blockIdx.x * blockDim.x + threadIdx.x;
    if (i >= n) return;
    float f = mono2f(em[i]);
    if (!isfinite(f)) f = 0.f;
    em[i] = __float_as_uint(f);
}

// ---------------------------------------------------------------------
// Edge pass 2 (layer 1): wave32 per edge. Recompute score, p=exp(sc-emax[dst]),
// atomicAdd denom (lanes 0..3) and p * xw1[src, h, lane] into acc1[dst].
__global__ void edge_aggr_l1(const int* __restrict__ src, const int* __restrict__ dst,
                             const float* __restrict__ eattr, const float* __restrict__ wea1,
                             const float* __restrict__ s_i1, const float* __restrict__ s_j1,
                             const float* __restrict__ emax1, float* __restrict__ denom1,
                             const float* __restrict__ xw1, float* __restrict__ acc1)
{
    int lane = threadIdx.x & 31;
    int wave = (blockIdx.x * blockDim.x + threadIdx.x) >> 5;
    int nw = (gridDim.x * blockDim.x) >> 5;
    for (int e = wave; e < EE; e += nw) {
        int s = src[e], d = dst[e];
        float se[4] = {0.f, 0.f, 0.f, 0.f};
#pragma unroll
        for (int k = 0; k < 16; ++k) {
            float ea = eattr[(size_t)e * 16 + k];
            se[0] += ea * wea1[k];      se[1] += ea * wea1[16 + k];
            se[2] += ea * wea1[32 + k]; se[3] += ea * wea1[48 + k];
        }
        float p[4];
#pragma unroll
        for (int h = 0; h < 4; ++h) {
            float sc = s_i1[d * 4 + h] + s_j1[s * 4 + h] + se[h];
            sc = (sc >= 0.f) ? sc : NEGS * sc;
            p[h] = expf(sc - emax1[d * 4 + h]);
        }
        if (lane < 4) atomicAdd(&denom1[d * 4 + lane], p[lane]);
#pragma unroll
        for (int h = 0; h < 4; ++h) {
            float v = p[h] * xw1[(size_t)s * 128 + h * 32 + lane];
            atomicAdd(&acc1[(size_t)d * 128 + h * 32 + lane], v);
        }
    }
}

__global__ void edge_aggr_l2(const int* __restrict__ src, const int* __restrict__ dst,
                             const float* __restrict__ eattr, const float* __restrict__ wea2,
                             const float* __restrict__ s_i2, const float* __restrict__ s_j2,
                             const float* __restrict__ emax2, float* __restrict__ denom2,
                             const float* __restrict__ hw2, float* __restrict__ acc2)
{
    int lane = threadIdx.x & 31;
    int wave = (blockIdx.x * blockDim.x + threadIdx.x) >> 5;
    int nw = (gridDim.x * blockDim.x) >> 5;
    for (int e = wave; e < EE; e += nw) {
        int s = src[e], d = dst[e];
        float se = 0.f;
#pragma unroll
        for (int k = 0; k < 16; ++k) se += eattr[(size_t)e * 16 + k] * wea2[k];
        float sc = s_i2[d] + s_j2[s] + se;
        sc = (sc >= 0.f) ? sc : NEGS * sc;
        float p = expf(sc - emax2[d]);
        if (lane == 0) atomicAdd(&denom2[d], p);
        atomicAdd(&acc2[(size_t)d * 32 + lane], p * hw2[(size_t)s * 32 + lane]);
    }
}

// ---------------------------------------------------------------------
// acc1 /= denom1 (per node,head) -> h_concat in place
__global__ void finalize_l1(float* __restrict__ acc1, const float* __restrict__ denom1)
{
    int i = blockIdx.x * blockDim.x + threadIdx.x;
    if (i >= NN * 128) return;
    int n = i >> 7, h = (i >> 5) & 3;
    float dnm = denom1[n * 4 + h];
    acc1[i] = (dnm > 0.f) ? acc1[i] / dnm : 0.f;
}

// out[n] = elu( (acc2[n,:]/denom2[n]) . proj2_w + proj2_b )
__global__ void final_out(const float* __restrict__ acc2, const float* __restrict__ denom2,
                          const float* __restrict__ p2w, const float* __restrict__ p2b,
                          float* __restrict__ out)
{
    int n = blockIdx.x * blockDim.x + threadIdx.x;
    if (n >= NN) return;
    float dnm = denom2[n];
    float v = 0.f;
#pragma unroll
    for (int o = 0; o < 32; ++o) {
        float hv = (dnm > 0.f) ? acc2[(size_t)n * 32 + o] / dnm : 0.f;
        v += hv * p2w[o];
    }
    v += p2b[0];
    out[n] = (v > 0.f) ? v : (expf(v) - 1.f);
}

// ---------------------------------------------------------------------
extern "C" void kernel_launch(void* const* d_in, const int* in_sizes, int n_in,
                              void* d_out, int out_size, void* d_ws, size_t ws_size,
                              hipStream_t stream)
{
    const float* x       = (const float*)d_in[0];
    const int*   eidx    = (const int*)d_in[1];
    const float* eattr   = (const float*)d_in[2];
    const float* W1      = (const float*)d_in[3];
    const float* We1     = (const float*)d_in[4];
    const float* att1    = (const float*)d_in[5];
    const float* proj1_w = (const float*)d_in[6];
    const float* proj1_b = (const float*)d_in[7];
    const float* W2      = (const float*)d_in[8];
    const float* We2     = (const float*)d_in[9];
    const float* att2    = (const float*)d_in[10];
    const float* proj2_w = (const float*)d_in[11];
    const float* proj2_b = (const float*)d_in[12];
    float* out = (float*)d_out;

    const int* srcp = eidx;
    const int* dstp = eidx + EE;

    // workspace layout (floats)
    float* ws     = (float*)d_ws;
    float* xw1    = ws;                          // N*128  (reused as h after proj1)
    float* acc1   = xw1 + (size_t)NN * 128;      // N*128
    float* s_i1   = acc1 + (size_t)NN * 128;     // N*4
    float* s_j1   = s_i1 + (size_t)NN * 4;       // N*4
    float* emax1  = s_j1 + (size_t)NN * 4;       // N*4 (uint view during max phase)
    float* denom1 = emax1 + (size_t)NN * 4;      // N*4
    float* hw2    = denom1 + (size_t)NN * 4;     // N*32
    float* acc2   = hw2 + (size_t)NN * 32;       // N*32
    float* s_i2   = acc2 + (size_t)NN * 32;      // N
    float* s_j2   = s_i2 + (size_t)NN;           // N
    float* emax2  = s_j2 + (size_t)NN;           // N
    float* denom2 = emax2 + (size_t)NN;          // N
    float* Wcat1  = denom2 + (size_t)NN;         // 2048
    float* wea1   = Wcat1 + 2048;                // 64
    float* wea2   = wea1 + 64;                   // 16

    prep_weights<<<1, 256, 0, stream>>>(W1, We1, att1, We2, att2, Wcat1, wea1, wea2);
    init_buffers<<<2048, 256, 0, stream>>>((unsigned*)emax1, denom1, acc1,
                                           (unsigned*)emax2, denom2, acc2);

    // Layer 1 node GEMM: xw1 = x @ Wcat1  (N x 16 @ 16 x 128)
    {
        int tiles = (NN / 16) * (128 / 16);
        wmma_gemm<<<(tiles + 7) / 8, 256, 0, stream>>>(x, 16, Wcat1, 128, xw1, 128,
                                                       NN, 128, 16, nullptr, 0);
    }
    scores1_kernel<<<(NN * 4 + 255) / 256, 256, 0, stream>>>(xw1, att1, s_i1, s_j1);
    edge_scores_l1<<<(EE * 4 + 255) / 256, 256, 0, stream>>>(srcp, dstp, eattr, wea1,
                                                             s_i1, s_j1, (unsigned*)emax1);
    decode_emax<<<(NN * 4 + 255) / 256, 256, 0, stream>>>((unsigned*)emax1, NN * 4);
    edge_aggr_l1<<<4096, 256, 0, stream>>>(srcp, dstp, eattr, wea1, s_i1, s_j1,
                                           emax1, denom1, xw1, acc1);
    finalize_l1<<<(NN * 128 + 255) / 256, 256, 0, stream>>>(acc1, denom1);

    // h = elu(elu(h_concat @ proj1_w + b))  (N x 128 @ 128 x 128) -> reuse xw1 buffer
    {
        int tiles = (NN / 16) * (128 / 16);
        wmma_gemm<<<(tiles + 7) / 8, 256, 0, stream>>>(acc1, 128, proj1_w, 128, xw1, 128,
                                                       NN, 128, 128, proj1_b, 2);
    }
    // hw2 = h @ W2  (N x 128 @ 128 x 32)
    {
        int tiles = (NN / 16) * (32 / 16);
        wmma_gemm<<<(tiles + 7) / 8, 256, 0, stream>>>(xw1, 128, W2, 32, hw2, 32,
                                                       NN, 32, 128, nullptr, 0);
    }
    scores2_kernel<<<(NN + 255) / 256, 256, 0, stream>>>(hw2, att2, s_i2, s_j2);
    edge_scores_l2<<<(EE + 255) / 256, 256, 0, stream>>>(srcp, dstp, eattr, wea2,
                                                         s_i2, s_j2, (unsigned*)emax2);
    decode_emax<<<(NN + 255) / 256, 256, 0, stream>>>((unsigned*)emax2, NN);
    edge_aggr_l2<<<4096, 256, 0, stream>>>(srcp, dstp, eattr, wea2, s_i2, s_j2,
                                           emax2, denom2, hw2, acc2);
    final_out<<<(NN + 255) / 256, 256, 0, stream>>>(acc2, denom2, proj2_w, proj2_b, out);
}